// MultiHeadAttentionLayer_73418170958210
// MI455X (gfx1250) — compile-verified
//
#include <hip/hip_runtime.h>
#include <hip/hip_bf16.h>

// Problem constants from the reference
#define N_NODES 50000
#define N_EDGES 800000
#define IN_DIM  64
#define NH      4      // heads
#define HD      16     // head dim
#define DC      3
#define HDIM    (NH * HD)   // 64

typedef __attribute__((ext_vector_type(2))) float v2f;
typedef __attribute__((ext_vector_type(8))) float v8f;

// ---------------------------------------------------------------------------
// Kernel 1: fused Q/K/V projection via V_WMMA_F32_16X16X4_F32.
// One wave (32 threads) computes one 16x16 fp32 output tile.
// grid = (N/16, 64/16, 3); blockIdx.z selects (Wq,bq,Q) / (Wk,bk,K) / (Wv,bv,V).
//
// VGPR layouts per CDNA5 ISA 7.12.2:
//   A (16x4 f32):  lane L holds h[row0 + L%16][k + 2*(L>>4) + {0,1}] in v0,v1
//   B (4x16 f32):  lane L holds W[k + 2*(L>>4) + {0,1}][col0 + L%16] in v0,v1
//   C/D (16x16):   lane L, vgpr j  ->  C[8*(L>>4) + j][L%16]
// ---------------------------------------------------------------------------
__global__ __launch_bounds__(32)
void qkv_proj_wmma(const float* __restrict__ h,
                   const float* __restrict__ Wq, const float* __restrict__ bq,
                   const float* __restrict__ Wk, const float* __restrict__ bk,
                   const float* __restrict__ Wv, const float* __restrict__ bv,
                   float* __restrict__ Q, float* __restrict__ K,
                   float* __restrict__ V)
{
    const float* W;  const float* bias;  float* out;
    if (blockIdx.z == 0)      { W = Wq; bias = bq; out = Q; }
    else if (blockIdx.z == 1) { W = Wk; bias = bk; out = K; }
    else                      { W = Wv; bias = bv; out = V; }

    const int row0 = blockIdx.x * 16;
    const int col0 = blockIdx.y * 16;
    const int lane = threadIdx.x;        // 0..31 (wave32)
    const int n    = lane & 15;          // column within tile / row within A
    const int hi   = lane >> 4;          // half-wave selector

    v8f acc = {};

    #pragma unroll
    for (int k = 0; k < IN_DIM; k += 4) {
        const int ka = k + 2 * hi;       // this lane's K pair base
        v2f a, b;
        // A: 16x4 slice of h (row-major [N, 64])
        a.x = h[(row0 + n) * IN_DIM + ka + 0];
        a.y = h[(row0 + n) * IN_DIM + ka + 1];
        // B: 4x16 slice of W (row-major [64, 64])
        b.x = W[(ka + 0) * HDIM + col0 + n];
        b.y = W[(ka + 1) * HDIM + col0 + n];
        acc = __builtin_amdgcn_wmma_f32_16x16x4_f32(
            /*neg_a=*/false, a, /*neg_b=*/false, b,
            /*c_mod=*/(short)0, acc, /*reuse_a=*/false, /*reuse_b=*/false);
    }

    const float bcol  = bias[col0 + n];
    const int   mbase = hi * 8;
    #pragma unroll
    for (int j = 0; j < 8; ++j)
        out[(row0 + mbase + j) * HDIM + col0 + n] = acc[j] + bcol;
}

// ---------------------------------------------------------------------------
// Kernel 2: edge phase. 16 lanes cooperate on one (edge, head) pair:
//   lane t: partial K[src]*Q[dst] dot at dim t; lanes 0..2 also hold the
//   s_l squared-distance partials. One butterfly reduces both sums.
//   Then: score = exp(clip(dot/4, -5, 5)); w = exp(-(dij^2)^2); news = score*w;
//   lane t: atomicAdd(wV[dst][h][t], V[src][h][t]*news); lane0: atomicAdd z.
// Consecutive 16-lane groups -> consecutive heads/edges, so each gather is a
// contiguous 64B request; Q/K/V/s_l/accumulators all live in the 192MB L2.
// ---------------------------------------------------------------------------
__global__ __launch_bounds__(256)
void edge_attention(const float* __restrict__ Q, const float* __restrict__ K,
                    const float* __restrict__ V, const float* __restrict__ s_l,
                    const int* __restrict__ src, const int* __restrict__ dst,
                    float* __restrict__ wV,   // [N, H, D] accumulator == d_out
                    float* __restrict__ z)    // [N, H]    accumulator
{
    const int t  = threadIdx.x & 15;                              // dim lane
    const int g  = blockIdx.x * (256 / 16) + (threadIdx.x >> 4);  // (edge,head)
    if (g >= N_EDGES * NH) return;

    const int e  = g >> 2;        // g / NH
    const int hd = g & 3;         // g % NH
    const int u  = src[e];
    const int v  = dst[e];

    const int qkOffU = u * HDIM + hd * HD + t;
    const int qkOffV = v * HDIM + hd * HD + t;

    float dotp = K[qkOffU] * Q[qkOffV];

    float dsq = 0.0f;
    if (t < DC) {
        const float d = s_l[u * (NH * DC) + hd * DC + t]
                      - s_l[v * (NH * DC) + hd * DC + t];
        dsq = d * d;
    }

    // Butterfly reduction across the 16-lane group (wave32: width arg = 16)
    #pragma unroll
    for (int m = 1; m < 16; m <<= 1) {
        dotp += __shfl_xor(dotp, m, 16);
        dsq  += __shfl_xor(dsq,  m, 16);
    }

    const float scaled = fminf(fmaxf(dotp * 0.25f, -5.0f), 5.0f); // /sqrt(16)
    const float score  = __expf(scaled);
    const float w      = __expf(-dsq * dsq);
    const float news   = score * w;

    atomicAdd(&wV[qkOffV], V[qkOffU] * news);
    if (t == 0) atomicAdd(&z[v * NH + hd], score);
}

// ---------------------------------------------------------------------------
// Kernel 3: finalize. out = (z > 0) ? wV / z : wV   (in place on d_out)
// ---------------------------------------------------------------------------
__global__ __launch_bounds__(256)
void finalize(float* __restrict__ out, const float* __restrict__ z)
{
    const int idx = blockIdx.x * 256 + threadIdx.x;
    if (idx >= N_NODES * HDIM) return;
    const int node = idx >> 6;               // / 64
    const int hd   = (idx >> 4) & 3;         // head
    const float zv = z[node * NH + hd];
    if (zv > 0.0f) out[idx] /= zv;
}

// ---------------------------------------------------------------------------
// Launcher. Input order: h, s_l, Wq, bq, Wk, bk, Wv, bv, src, dst
// Workspace: Q | K | V | z   (3*N*64 + N*4 floats ~ 39.2 MB)
// ---------------------------------------------------------------------------
extern "C" void kernel_launch(void* const* d_in, const int* in_sizes, int n_in,
                              void* d_out, int out_size, void* d_ws, size_t ws_size,
                              hipStream_t stream)
{
    const float* h   = (const float*)d_in[0];
    const float* s_l = (const float*)d_in[1];
    const float* Wq  = (const float*)d_in[2];
    const float* bq  = (const float*)d_in[3];
    const float* Wk  = (const float*)d_in[4];
    const float* bk  = (const float*)d_in[5];
    const float* Wv  = (const float*)d_in[6];
    const float* bv  = (const float*)d_in[7];
    const int*   src = (const int*)d_in[8];
    const int*   dst = (const int*)d_in[9];

    float* ws = (float*)d_ws;
    float* Q  = ws;
    float* K  = Q + (size_t)N_NODES * HDIM;
    float* V  = K + (size_t)N_NODES * HDIM;
    float* z  = V + (size_t)N_NODES * HDIM;

    float* out = (float*)d_out;   // [N, H, D] fp32

    // Zero the accumulators (harness poisons d_out / d_ws).
    hipMemsetAsync(out, 0, (size_t)N_NODES * HDIM * sizeof(float), stream);
    hipMemsetAsync(z,   0, (size_t)N_NODES * NH   * sizeof(float), stream);

    // 1) QKV projection: one wave per 16x16 tile, z-dim picks Q/K/V.
    dim3 gGemm(N_NODES / 16, HDIM / 16, 3);
    qkv_proj_wmma<<<gGemm, 32, 0, stream>>>(h, Wq, bq, Wk, bk, Wv, bv, Q, K, V);

    // 2) Edge phase: 16 lanes per (edge, head); 16 groups per 256-thread block.
    const int groups = N_EDGES * NH;
    dim3 gEdge((groups + 15) / 16, 1, 1);
    edge_attention<<<gEdge, 256, 0, stream>>>(Q, K, V, s_l, src, dst, out, z);

    // 3) Finalize.
    const int total = N_NODES * HDIM;
    finalize<<<(total + 255) / 256, 256, 0, stream>>>(out, z);
}